// MultiHeadedAttention_73246372266542
// MI455X (gfx1250) — compile-verified
//
#include <hip/hip_runtime.h>
#include <hip/hip_bf16.h>
#include <stdint.h>

// MI455X / gfx1250, wave32. fp32 WMMA (V_WMMA_F32_16X16X4_F32) for all GEMMs,
// double-buffered async global->LDS staging (GLOBAL_LOAD_ASYNC_TO_LDS_B128,
// s_wait_asynccnt) overlapping copy of chunk i+1 with compute of chunk i.
typedef float v2f __attribute__((ext_vector_type(2)));
typedef float v8f __attribute__((ext_vector_type(8)));

#define MHA_B 2
#define MHA_S 2048
#define MHA_E 1024
#define MHA_H 16
#define MHA_D 64
#define MHA_M (MHA_B * MHA_S)   // 4096
#define LDSS 36                  // padded LDS row stride (floats): 16B-aligned rows, conflict-free

// LDS byte offset of a shared-memory pointer: per ISA, the LDS aperture keeps
// the LDS offset in addr[31:0], so truncating the generic pointer is exact.
__device__ __forceinline__ uint32_t lds_off(const void* p) {
  return (uint32_t)(uintptr_t)p;
}
__device__ __forceinline__ void async_load_b128(uint32_t lds, const void* gptr) {
  unsigned long long ga = (unsigned long long)(uintptr_t)gptr;
  asm volatile("global_load_async_to_lds_b128 %0, %1, off"
               :: "v"(lds), "v"(ga) : "memory");
}
template <int N>
__device__ __forceinline__ void wait_asynccnt() {
  asm volatile("s_wait_asynccnt %0" :: "i"(N) : "memory");
}

// ---------------------------------------------------------------------------
// NT GEMM: out[m,n] = sum_k A[m,k] * Bw[n,k]  (+ optional per-(m,head(k)) scale
// of A, + optional residual add). Block: 128 threads (4 waves).
// Block tile 64x128, wave tile 32x64 (2x4 WMMA tiles), K chunk 32 via LDS.
// Non-scaled path: double-buffered async LDS staging (12 b128 loads/thread per
// chunk; async loads complete in order, so s_wait_asynccnt 12 == chunk landed).
// ---------------------------------------------------------------------------
template <bool SCALE_A, bool ADD_SKIP>
__global__ __launch_bounds__(128) void gemm_nt_wmma(
    const float* __restrict__ A, const float* __restrict__ Bw,
    const float* __restrict__ coeff, const float* __restrict__ skip,
    float* __restrict__ out, int M, int N, int K) {
  __shared__ float As[2][64 * LDSS];
  __shared__ float Bs[2][128 * LDSS];

  const int mBase = blockIdx.y * 64;
  const int nBase = blockIdx.x * 128;
  const int t = threadIdx.x;
  const int lane = t & 31;
  const int w = t >> 5;
  const int mw = (w & 1) * 32;   // wave's m offset in block tile
  const int nw = (w >> 1) * 64;  // wave's n offset in block tile
  const int lh = lane & 15;
  const int kq = (lane >> 4) * 2;  // fragment K sub-offset (ISA 16x4 layout)

  v8f acc[2][4] = {};

  // Issue this thread's 12 async b128 loads (4 for A, 8 for B) for one chunk.
  auto stage_async = [&](int buf, int kc) {
#pragma unroll
    for (int i = 0; i < 4; ++i) {
      const int idx = t + i * 128;        // A: 64 rows x 8 float4
      const int row = idx >> 3;
      const int c4 = (idx & 7) * 4;
      async_load_b128(lds_off(&As[buf][row * LDSS + c4]),
                      &A[(size_t)(mBase + row) * K + kc + c4]);
    }
#pragma unroll
    for (int i = 0; i < 8; ++i) {
      const int idx = t + i * 128;        // B: 128 rows x 8 float4
      const int row = idx >> 3;
      const int c4 = (idx & 7) * 4;
      async_load_b128(lds_off(&Bs[buf][row * LDSS + c4]),
                      &Bw[(size_t)(nBase + row) * K + kc + c4]);
    }
  };

  auto compute_chunk = [&](int buf) {
    const float* Ab = &As[buf][0];
    const float* Bb = &Bs[buf][0];
#pragma unroll
    for (int k0 = 0; k0 < 32; k0 += 4) {
      v2f a0 = *(const v2f*)&Ab[(mw + lh) * LDSS + k0 + kq];
      v2f a1 = *(const v2f*)&Ab[(mw + 16 + lh) * LDSS + k0 + kq];
      v2f bfr[4];
#pragma unroll
      for (int j = 0; j < 4; ++j)
        bfr[j] = *(const v2f*)&Bb[(nw + 16 * j + lh) * LDSS + k0 + kq];
#pragma unroll
      for (int j = 0; j < 4; ++j) {
        acc[0][j] = __builtin_amdgcn_wmma_f32_16x16x4_f32(
            false, a0, false, bfr[j], (short)0, acc[0][j], false, false);
        acc[1][j] = __builtin_amdgcn_wmma_f32_16x16x4_f32(
            false, a1, false, bfr[j], (short)0, acc[1][j], false, false);
      }
    }
  };

  if constexpr (SCALE_A) {
    // Sync staging with per-(row, head) scaling of A. A K-chunk of 32 never
    // straddles a head boundary (32 | 64), so one coeff per float4.
    for (int kc = 0; kc < K; kc += 32) {
      __syncthreads();
#pragma unroll
      for (int i = 0; i < 4; ++i) {
        const int idx = t + i * 128;        // A 64x32
        const int row = idx >> 3;
        const int c4 = (idx & 7) * 4;
        float4 av = *(const float4*)&A[(size_t)(mBase + row) * K + kc + c4];
        const int m = mBase + row;
        const int b = m >> 11;              // S = 2048
        const int s = m & 2047;
        const int h = kc >> 6;              // head of this K-chunk (D = 64)
        const float cf = coeff[(size_t)(b * MHA_H + h) * MHA_S + s];
        av.x *= cf; av.y *= cf; av.z *= cf; av.w *= cf;
        *(float4*)&As[0][row * LDSS + c4] = av;
      }
#pragma unroll
      for (int i = 0; i < 8; ++i) {
        const int idx = t + i * 128;        // B 128x32
        const int row = idx >> 3;
        const int c4 = (idx & 7) * 4;
        *(float4*)&Bs[0][row * LDSS + c4] =
            *(const float4*)&Bw[(size_t)(nBase + row) * K + kc + c4];
      }
      __syncthreads();
      compute_chunk(0);
    }
  } else {
    // Software pipeline: copy chunk i+1 (async) while computing chunk i.
    const int nCh = K >> 5;
    stage_async(0, 0);
    for (int ic = 0; ic < nCh; ++ic) {
      if (ic + 1 < nCh) {
        stage_async((ic + 1) & 1, (ic + 1) << 5);
        wait_asynccnt<12>();  // chunk ic landed; ic+1 still streaming
      } else {
        wait_asynccnt<0>();
      }
      __syncthreads();        // all waves' chunk-ic loads complete
      compute_chunk(ic & 1);
      __syncthreads();        // buffer ic&1 free for reuse at iteration ic+1
    }
  }

  // Epilogue. C layout: VGPR r -> row r (lanes 0-15) / row r+8 (lanes 16-31),
  // column = lane & 15.
  const int mh = (lane >> 4) * 8;
#pragma unroll
  for (int i = 0; i < 2; ++i)
#pragma unroll
    for (int j = 0; j < 4; ++j)
#pragma unroll
      for (int r = 0; r < 8; ++r) {
        const int m = mBase + mw + 16 * i + mh + r;
        const int n = nBase + nw + 16 * j + lh;
        float val = acc[i][j][r];
        if (ADD_SKIP) val += skip[(size_t)m * N + n];
        out[(size_t)m * N + n] = val;
      }
}

// ---------------------------------------------------------------------------
// Causal softmax diagonal coefficient: coeff[b,h,s] = p[s,s] where
// p = softmax_t((q_s . k_t)/32, t <= s). One wave per 16 rows of one (b,h).
// Scores via WMMA f32 16x16x4 over D=64: two independent 8-deep chains.
// ---------------------------------------------------------------------------
__global__ __launch_bounds__(32) void attn_diag_coeff(
    const float* __restrict__ q, const float* __restrict__ k,
    float* __restrict__ coeff) {
  const int sblk = blockIdx.x;   // 0..S/16-1
  const int bh = blockIdx.y;     // 0..B*H-1
  const int b = bh >> 4;
  const int h = bh & 15;
  const int lane = threadIdx.x;
  const int lh = lane & 15;
  const int kq = (lane >> 4) * 2;
  const int s0 = sblk * 16;
  const float invNorm = 1.0f / 32.0f;  // 1/sqrt(1024)

  // Q fragments for rows s0..s0+15, D=64 -> 16 WMMA K-steps, kept in VGPRs.
  const float* qrow = q + (size_t)(b * MHA_S + s0 + lh) * MHA_E + h * MHA_D;
  v2f afrag[16];
#pragma unroll
  for (int kk = 0; kk < 16; ++kk) afrag[kk] = *(const v2f*)(qrow + kk * 4 + kq);

  float pm[8], ps[8];
  float ds = 0.0f;  // diagonal logit for the row this lane owns (if any)
#pragma unroll
  for (int r = 0; r < 8; ++r) { pm[r] = -3.0e38f; ps[r] = 0.0f; }
  const int myM = (lane < 16) ? 0 : 8;

  for (int jt = 0; jt <= sblk; ++jt) {
    const float* krow =
        k + (size_t)(b * MHA_S + jt * 16 + lh) * MHA_E + h * MHA_D;
    v2f bfrag[16];
#pragma unroll
    for (int kk = 0; kk < 16; ++kk)
      bfrag[kk] = *(const v2f*)(krow + kk * 4 + kq);
    v8f c0 = {}, c1 = {};
#pragma unroll
    for (int kk = 0; kk < 8; ++kk) {  // two interleaved accumulation chains
      c0 = __builtin_amdgcn_wmma_f32_16x16x4_f32(false, afrag[kk], false,
                                                 bfrag[kk], (short)0, c0,
                                                 false, false);
      c1 = __builtin_amdgcn_wmma_f32_16x16x4_f32(false, afrag[kk + 8], false,
                                                 bfrag[kk + 8], (short)0, c1,
                                                 false, false);
    }
    const v8f c = c0 + c1;
    const int n = lh;  // column within tile for this lane
    const bool fullTile = (jt < sblk);
#pragma unroll
    for (int r = 0; r < 8; ++r) {
      const int m = myM + r;
      const float val = c[r] * invNorm;
      if (!fullTile && n == m) ds = val;  // diagonal tile: t0 == s0
      if (fullTile || (n <= m)) {         // causal mask t <= s
        const float nm = fmaxf(pm[r], val);
        ps[r] = ps[r] * __expf(pm[r] - nm) + __expf(val - nm);
        pm[r] = nm;
      }
    }
  }

  // Reduce (max,sum) across the 16 lanes of each half (xor masks stay in half).
#pragma unroll
  for (int off = 1; off < 16; off <<= 1) {
#pragma unroll
    for (int r = 0; r < 8; ++r) {
      const float om = __shfl_xor(pm[r], off, 32);
      const float os = __shfl_xor(ps[r], off, 32);
      const float nm = fmaxf(pm[r], om);
      ps[r] = ps[r] * __expf(pm[r] - nm) + os * __expf(om - nm);
      pm[r] = nm;
    }
  }

  // Diagonal owner lanes: 0..7 -> rows 0..7 (r=lane); 24..31 -> rows 8..15.
  int row = -1, r = 0;
  if (lane < 8) { row = lane; r = lane; }
  else if (lane >= 24) { row = lane - 16; r = lane - 24; }
  if (row >= 0)
    coeff[(size_t)bh * MHA_S + s0 + row] = __expf(ds - pm[r]) / ps[r];
}

// ---------------------------------------------------------------------------
extern "C" void kernel_launch(void* const* d_in, const int* in_sizes, int n_in,
                              void* d_out, int out_size, void* d_ws,
                              size_t ws_size, hipStream_t stream) {
  const float* x  = (const float*)d_in[0];
  const float* Wq = (const float*)d_in[1];
  const float* Wk = (const float*)d_in[2];
  const float* Wv = (const float*)d_in[3];
  const float* Wo = (const float*)d_in[4];

  const size_t MN = (size_t)MHA_M * MHA_E;  // 4096*1024
  float* q = (float*)d_ws;
  float* k = q + MN;
  float* v = k + MN;
  float* coeff = v + MN;  // B*H*S floats

  dim3 gGemm(MHA_E / 128, MHA_M / 64);  // (8, 64)
  dim3 bGemm(128);

  // Q/K/V projections: x @ W^T  (NT GEMM, pipelined async LDS staging).
  gemm_nt_wmma<false, false><<<gGemm, bGemm, 0, stream>>>(
      x, Wq, nullptr, nullptr, q, MHA_M, MHA_E, MHA_E);
  gemm_nt_wmma<false, false><<<gGemm, bGemm, 0, stream>>>(
      x, Wk, nullptr, nullptr, k, MHA_M, MHA_E, MHA_E);
  gemm_nt_wmma<false, false><<<gGemm, bGemm, 0, stream>>>(
      x, Wv, nullptr, nullptr, v, MHA_M, MHA_E, MHA_E);

  // Diagonal softmax coefficients.
  dim3 gAttn(MHA_S / 16, MHA_B * MHA_H);  // (128, 32)
  attn_diag_coeff<<<gAttn, dim3(32), 0, stream>>>(q, k, coeff);

  // out = (coeff ⊙ v) @ Wo^T + v
  gemm_nt_wmma<true, true><<<gGemm, bGemm, 0, stream>>>(
      v, Wo, coeff, v, (float*)d_out, MHA_M, MHA_E, MHA_E);
}